// DigitsCapsule_9947144257640
// MI455X (gfx1250) — compile-verified
//
#include <hip/hip_runtime.h>
#include <math.h>

// Problem sizes (fixed by the reference)
#define B_TOT 512
#define I_TOT 1152
#define K_DIM 8
#define L_DIM 16
#define O_DIM 7
#define N_DIM 112   // L_DIM * O_DIM
#define NT    7     // N tiles of 16

typedef float v2f __attribute__((ext_vector_type(2)));
typedef float v8f __attribute__((ext_vector_type(8)));
typedef int   v4i __attribute__((ext_vector_type(4)));

// gfx1250 async global->LDS path (ASYNCcnt), with vectorized fallback
#if defined(__has_builtin)
# if __has_builtin(__builtin_amdgcn_global_load_async_to_lds_b128) && \
     __has_builtin(__builtin_amdgcn_s_wait_asynccnt)
#  define HAVE_ASYNC_LDS 1
# endif
#endif
#ifndef HAVE_ASYNC_LDS
# define HAVE_ASYNC_LDS 0
#endif

// Builtin parameter types (from hipcc diagnostic): (int4 AS1*, int4 AS3*, Ii, Ii)
#define ASYNC_GPTR(p) ((__attribute__((address_space(1))) v4i*)(p))
#define ASYNC_LPTR(p) ((__attribute__((address_space(3))) v4i*)(p))

// D = A(16x8) * B(8x112-tile) via two chained V_WMMA_F32_16X16X4_F32
__device__ __forceinline__ v8f wmma_k8(v2f a0, v2f a1, v2f b0, v2f b1) {
  v8f d = {0.f, 0.f, 0.f, 0.f, 0.f, 0.f, 0.f, 0.f};
  d = __builtin_amdgcn_wmma_f32_16x16x4_f32(false, a0, false, b0, (short)0, d, false, false);
  d = __builtin_amdgcn_wmma_f32_16x16x4_f32(false, a1, false, b1, (short)0, d, false, false);
  return d;
}

// ---------------- zero b_ij (must run every call: ws is not re-poisoned) ----
__global__ void caps_zero_b(float* __restrict__ bij) {
  int idx = blockIdx.x * 256 + threadIdx.x;
  if (idx < I_TOT * L_DIM) bij[idx] = 0.f;
}

// ---------------- c = softmax(b_ij, axis=I)  [1152,16], one block ----------
__global__ void __launch_bounds__(256) caps_softmax(const float* __restrict__ bij,
                                                    float* __restrict__ c) {
  __shared__ float red[256];
  __shared__ float colmax[16];
  __shared__ float colsum[16];
  int tid = threadIdx.x;
  int l = tid & 15;
  int chunk = tid >> 4;           // 16 chunks of 72 rows
  int i0 = chunk * 72;

  float m = -3.402823466e38f;
  for (int r = 0; r < 72; ++r) m = fmaxf(m, bij[(i0 + r) * L_DIM + l]);
  red[tid] = m;
  __syncthreads();
  if (tid < 16) {
    float mm = red[tid];
    for (int cc = 1; cc < 16; ++cc) mm = fmaxf(mm, red[cc * 16 + tid]);
    colmax[tid] = mm;
  }
  __syncthreads();
  float mm = colmax[l];
  float s = 0.f;
  for (int r = 0; r < 72; ++r) s += expf(bij[(i0 + r) * L_DIM + l] - mm);
  __syncthreads();
  red[tid] = s;
  __syncthreads();
  if (tid < 16) {
    float tot = 0.f;
    for (int cc = 0; cc < 16; ++cc) tot += red[cc * 16 + tid];
    colsum[tid] = tot;
  }
  __syncthreads();
  float denom = colsum[l];
  for (int r = 0; r < 72; ++r) {
    int idx = (i0 + r) * L_DIM + l;
    c[idx] = expf(bij[idx] - mm) / denom;
  }
}

// ---- s_j = sum_i c[i,l]*u_hat (u_hat recomputed via WMMA), fused squash ----
// grid = 32 (b-tiles of 16), block = 256 (8 waves split the i-dimension)
__global__ void __launch_bounds__(256) caps_route_s(const float* __restrict__ x,
                                                    const float* __restrict__ w,
                                                    const float* __restrict__ c,
                                                    float* __restrict__ vout) {
  __shared__ float s_lds[16 * N_DIM];   // 7 KB
  int tid = threadIdx.x;
  int wave = tid >> 5, lane = tid & 31;
  int row = lane & 15, khalf = lane >> 4;
  int b0 = blockIdx.x * 16;

  for (int idx = tid; idx < 16 * N_DIM; idx += 256) s_lds[idx] = 0.f;
  __syncthreads();

  float sacc[NT][8];
#pragma unroll
  for (int t = 0; t < NT; ++t)
#pragma unroll
    for (int j = 0; j < 8; ++j) sacc[t][j] = 0.f;

  int kr = khalf * 2;
  for (int i = wave; i < I_TOT; i += 8) {   // uniform per wave -> EXEC all ones
    const float* xrow = x + ((size_t)(b0 + row) * I_TOT + i) * K_DIM;
    __builtin_prefetch(xrow + 8 * K_DIM, 0, 0);      // next i for this wave
    v2f a0 = *(const v2f*)(xrow + kr);               // K = kr, kr+1
    v2f a1 = *(const v2f*)(xrow + kr + 4);           // K = kr+4, kr+5
    const float* wrow = w + (size_t)i * (K_DIM * N_DIM);
#pragma unroll
    for (int t = 0; t < NT; ++t) {
      int col = t * 16 + row;
      v2f bw0, bw1;
      bw0.x = wrow[(kr + 0) * N_DIM + col];
      bw0.y = wrow[(kr + 1) * N_DIM + col];
      bw1.x = wrow[(kr + 4) * N_DIM + col];
      bw1.y = wrow[(kr + 5) * N_DIM + col];
      v8f d = wmma_k8(a0, a1, bw0, bw1);
      float cv = c[i * L_DIM + (col / 7)];           // same N for all 8 comps
#pragma unroll
      for (int j = 0; j < 8; ++j) sacc[t][j] += cv * d[j];
    }
  }

#pragma unroll
  for (int t = 0; t < NT; ++t) {
    int n = t * 16 + row;
#pragma unroll
    for (int j = 0; j < 8; ++j) {
      int M = j + khalf * 8;
      atomicAdd(&s_lds[M * N_DIM + n], sacc[t][j]);  // ds_add_f32 reduce
    }
  }
  __syncthreads();

  // squash over O, write v[b, o, l]
  int m = tid >> 4, l = tid & 15;
  float sv[O_DIM];
  float sq = 0.f;
#pragma unroll
  for (int o = 0; o < O_DIM; ++o) {
    sv[o] = s_lds[m * N_DIM + l * 7 + o];
    sq += sv[o] * sv[o];
  }
  float scale = sqrtf(sq) / (1.f + sq);   // (sq/(1+sq)) * (1/||s||)
  int b = b0 + m;
#pragma unroll
  for (int o = 0; o < O_DIM; ++o)
    vout[((size_t)(b * O_DIM + o)) * L_DIM + l] = sv[o] * scale;
}

// ---- b_ij += mean_B sum_O v[b,o,l]*u_hat[b,i,l,o] (u_hat via WMMA) --------
// grid = 144, block = 256 (8 waves, one i per wave); dynamic LDS caches v
__global__ void __launch_bounds__(256) caps_route_uv(const float* __restrict__ x,
                                                     const float* __restrict__ w,
                                                     const float* __restrict__ v,
                                                     float* __restrict__ bij) {
  extern __shared__ float smem[];
  float* v_lds = smem;                      // 512*112 floats = 229 KB
  float* uvn   = smem + B_TOT * N_DIM;      // 8*112 floats

  int tid = threadIdx.x;
  // Fill the 229 KB LDS v-cache: async global->LDS B128 (no VGPR staging),
  // or vectorized float4 copy if the builtin is unavailable.
  for (int idx4 = tid; idx4 < (B_TOT * N_DIM) / 4; idx4 += 256) {
#if HAVE_ASYNC_LDS
    __builtin_amdgcn_global_load_async_to_lds_b128(
        ASYNC_GPTR(v + (size_t)idx4 * 4),
        ASYNC_LPTR(v_lds + (size_t)idx4 * 4),
        0, 0);
#else
    *(float4*)(v_lds + (size_t)idx4 * 4) = *(const float4*)(v + (size_t)idx4 * 4);
#endif
  }
#if HAVE_ASYNC_LDS
  __builtin_amdgcn_s_wait_asynccnt(0);   // own async loads done before barrier
#endif
  __syncthreads();

  int wave = tid >> 5, lane = tid & 31;
  int row = lane & 15, khalf = lane >> 4;
  int i = blockIdx.x * 8 + wave;
  int kr = khalf * 2;

  // B operands are invariant across the b-loop: hoist all 7 n-tiles
  const float* wrow = w + (size_t)i * (K_DIM * N_DIM);
  v2f bw0[NT], bw1[NT];
#pragma unroll
  for (int t = 0; t < NT; ++t) {
    int col = t * 16 + row;
    v2f t0, t1;
    t0.x = wrow[(kr + 0) * N_DIM + col];
    t0.y = wrow[(kr + 1) * N_DIM + col];
    t1.x = wrow[(kr + 4) * N_DIM + col];
    t1.y = wrow[(kr + 5) * N_DIM + col];
    bw0[t] = t0; bw1[t] = t1;
  }

  float part[NT];
#pragma unroll
  for (int t = 0; t < NT; ++t) part[t] = 0.f;

  for (int bt = 0; bt < 32; ++bt) {
    int b0 = bt * 16;
    const float* xrow = x + ((size_t)(b0 + row) * I_TOT + i) * K_DIM;
    v2f a0 = *(const v2f*)(xrow + kr);
    v2f a1 = *(const v2f*)(xrow + kr + 4);
#pragma unroll
    for (int t = 0; t < NT; ++t) {
      v8f d = wmma_k8(a0, a1, bw0[t], bw1[t]);
      int n = t * 16 + row;
      int base = (n % 7) * L_DIM + (n / 7);          // v[b,o,l] -> b*112+base
#pragma unroll
      for (int j = 0; j < 8; ++j) {
        int b = b0 + j + khalf * 8;
        part[t] += v_lds[b * N_DIM + base] * d[j];
      }
    }
  }

  // fold the two half-waves (same N), then fold over O via LDS
#pragma unroll
  for (int t = 0; t < NT; ++t) part[t] += __shfl_xor(part[t], 16, 32);
  if (lane < 16) {
#pragma unroll
    for (int t = 0; t < NT; ++t) uvn[wave * N_DIM + t * 16 + lane] = part[t];
  }
  __syncthreads();
  if (lane < 16) {
    int l = lane;
    float uv = 0.f;
#pragma unroll
    for (int o = 0; o < O_DIM; ++o) uv += uvn[wave * N_DIM + l * 7 + o];
    bij[i * L_DIM + l] += uv * (1.f / (float)B_TOT);  // unique owner, no atomic
  }
}

// ---------------------------------------------------------------------------
extern "C" void kernel_launch(void* const* d_in, const int* in_sizes, int n_in,
                              void* d_out, int out_size, void* d_ws, size_t ws_size,
                              hipStream_t stream) {
  const float* x = (const float*)d_in[0];   // [512,1152,8]
  const float* w = (const float*)d_in[1];   // [1152,8,16,7]

  float* bij  = (float*)d_ws;               // [1152,16]
  float* cbuf = bij + I_TOT * L_DIM;        // [1152,16]
  float* vbuf = cbuf + I_TOT * L_DIM;       // [512,7,16]

  size_t vshm = (size_t)(B_TOT * N_DIM + 8 * N_DIM) * sizeof(float); // ~233 KB
  (void)hipFuncSetAttribute((const void*)caps_route_uv,
                            hipFuncAttributeMaxDynamicSharedMemorySize,
                            (int)vshm);

  caps_zero_b<<<(I_TOT * L_DIM + 255) / 256, 256, 0, stream>>>(bij);

  for (int it = 0; it < 3; ++it) {
    caps_softmax<<<1, 256, 0, stream>>>(bij, cbuf);
    float* vo = (it == 2) ? (float*)d_out : vbuf;
    caps_route_s<<<32, 256, 0, stream>>>(x, w, cbuf, vo);
    if (it < 2)
      caps_route_uv<<<144, 256, vshm, stream>>>(x, w, vbuf, bij);
  }
}